// HierarchicalSoftmax_91207925498218
// MI455X (gfx1250) — compile-verified
//
#include <hip/hip_runtime.h>
#include <hip/hip_bf16.h>
#include <math.h>

typedef __attribute__((ext_vector_type(2))) float v2f;
typedef __attribute__((ext_vector_type(8))) float v8f;

#define EMBED 128
#define WAVES_PER_BLOCK 8
#define HSVOCAB 50257

__global__ void hs_init(float* acc) {
    acc[0] = 0.0f;
    acc[1] = 0.0f;
}

__global__ __launch_bounds__(256) void hs_main(
    const float* __restrict__ emb,    // [T, 128]
    const int*   __restrict__ target, // [T]
    const float* __restrict__ fc,     // [V-1, 128]
    const int*   __restrict__ pidx,   // [V, D]
    const float* __restrict__ pcode,  // [V, D]
    const float* __restrict__ pmask,  // [V, D]
    float* __restrict__ acc,          // [2] : {sum_bce, sum_mask}
    int T, int D)
{
    __shared__ float lds_emb[WAVES_PER_BLOCK][EMBED];
    const int lane = threadIdx.x & 31;
    const int wave = threadIdx.x >> 5;
    const int t = blockIdx.x * WAVES_PER_BLOCK + wave;
    if (t >= T) return;   // wave-uniform exit; EXEC stays all-ones for WMMA

    // Stage this token's embedding row into LDS: 32 lanes x 4 floats = 128.
    {
        const float4* src = (const float4*)(emb + (size_t)t * EMBED);
        float4 v = src[lane];
        ((float4*)(&lds_emb[wave][0]))[lane] = v;
        // same-wave LDS ops are in-order; compiler inserts s_wait_dscnt for the reads below
    }

    const int tv = target[t];
    const size_t prow = (size_t)tv * (size_t)D;

    float sbce = 0.0f;
    float smask = 0.0f;

    // K-half handled by this lane (A layout: lanes 0-15 -> K=0,1 ; lanes 16-31 -> K=2,3)
    const int koff = (lane >> 4) << 1;  // 0 or 2
    const float* brow = &lds_emb[wave][koff];

    const int nchunks = (D + 15) >> 4;
    for (int dc = 0; dc < nchunks; ++dc) {
        // Depth handled by this lane's A row (M = lane&15 within chunk)
        const int dloc = dc * 16 + (lane & 15);
        const int dcl  = (dloc < D) ? dloc : (D - 1);
        const int   ridx = pidx[prow + dcl];
        const float code = pcode[prow + dcl];
        const float mask = (dloc < D) ? pmask[prow + dcl] : 0.0f;

        const float* arow = fc + (size_t)ridx * EMBED + koff;

        v8f c = {0.f, 0.f, 0.f, 0.f, 0.f, 0.f, 0.f, 0.f};
        // C[d][n] = sum_k fc[idx_d][k] * emb[t][k], replicated across n
        #pragma unroll
        for (int kb = 0; kb < EMBED; kb += 4) {
            v2f a = *(const v2f*)(arow + kb);  // A: 16x4 f32 tile (per-depth fc rows)
            v2f b = *(const v2f*)(brow + kb);  // B: 4x16 f32 tile (emb chunk, all cols equal)
            c = __builtin_amdgcn_wmma_f32_16x16x4_f32(
                /*neg_a=*/false, a, /*neg_b=*/false, b,
                /*c_mod=*/(short)0, c, /*reuse_a=*/false, /*reuse_b=*/false);
        }

        // C VGPR i holds M=i (lanes 0-15) / M=i+8 (lanes 16-31); value equal across N lanes.
        const int base = (lane & 16) ? 8 : 0;
        #pragma unroll
        for (int i = 0; i < 8; ++i) {
            float l  = c[i];
            float cd = __shfl(code, base + i, 32);  // lane (base+i) holds depth dc*16+base+i
            float md = __shfl(mask, base + i, 32);
            // stable BCE-with-logits: max(l,0) - l*code + log1p(exp(-|l|))
            float bce = fmaxf(l, 0.0f) - l * cd + log1pf(expf(-fabsf(l)));
            sbce  += bce * md;
            smask += md;
        }
    }

    // lanes 0-15 replicate one partial, lanes 16-31 the other.
    float tb = __shfl(sbce, 0, 32)  + __shfl(sbce, 16, 32);
    float tm = __shfl(smask, 0, 32) + __shfl(smask, 16, 32);
    if (lane == 0) {
        atomicAdd(&acc[0], tb);
        atomicAdd(&acc[1], tm);
    }
}

__global__ void hs_final(const float* __restrict__ acc, float* __restrict__ out) {
    out[0] = acc[0] / acc[1];
}

extern "C" void kernel_launch(void* const* d_in, const int* in_sizes, int n_in,
                              void* d_out, int out_size, void* d_ws, size_t ws_size,
                              hipStream_t stream) {
    const float* emb    = (const float*)d_in[0];  // [128,1,256,128] f32
    const int*   target = (const int*)  d_in[1];  // [128,1,256] i32
    const float* fc     = (const float*)d_in[2];  // [50256,128] f32
    const int*   pidx   = (const int*)  d_in[3];  // [50257,D] i32
    const float* pcode  = (const float*)d_in[4];  // [50257,D] f32
    const float* pmask  = (const float*)d_in[5];  // [50257,D] f32

    const int T = in_sizes[1];              // 32768 tokens
    const int D = in_sizes[3] / HSVOCAB;    // padded Huffman path depth

    float* acc = (float*)d_ws;

    hs_init<<<1, 1, 0, stream>>>(acc);

    const int blocks = (T + WAVES_PER_BLOCK - 1) / WAVES_PER_BLOCK;
    hs_main<<<blocks, 256, 0, stream>>>(emb, target, fc, pidx, pcode, pmask, acc, T, D);

    hs_final<<<1, 1, 0, stream>>>(acc, (float*)d_out);
}